// _GTN_39230231282113
// MI455X (gfx1250) — compile-verified
//
#include <hip/hip_runtime.h>
#include <hip/hip_bf16.h>
#include <math.h>

// ---------------------------------------------------------------------------
// Problem constants (from the reference)
// ---------------------------------------------------------------------------
#define NNODES   40000
#define NEDGES   640000
#define ETOT     (NEDGES + NNODES)   // edges + self loops = 680000
#define CH       128
#define OUTC     16
#define NGRAPH   64
#define EPS_F    1e-16f

typedef __attribute__((ext_vector_type(2))) float v2f;
typedef __attribute__((ext_vector_type(8))) float v8f;

// ---------------------------------------------------------------------------
// Helpers
// ---------------------------------------------------------------------------
__device__ inline void edge_nodes(int e, const int* __restrict__ ei, int& u, int& v) {
    if (e < NEDGES) { u = ei[e]; v = ei[NEDGES + e]; }
    else            { u = e - NEDGES; v = u; }          // self loop
}

__device__ inline void atomicMaxFloat(float* addr, float val) {
    unsigned int* ua = (unsigned int*)addr;
    unsigned int old = *ua;
    while (true) {
        float cur = __uint_as_float(old);
        if (cur >= val) break;
        unsigned int assumed = old;
        old = atomicCAS(ua, assumed, __float_as_uint(val));
        if (old == assumed) break;
    }
}

// ---------------------------------------------------------------------------
// fill kernel (workspace init; harness does not re-zero between replays)
// ---------------------------------------------------------------------------
__global__ void fill_kernel(float* __restrict__ p, float v, long n) {
    long i = (long)blockIdx.x * blockDim.x + threadIdx.x;
    if (i < n) p[i] = v;
}

// ---------------------------------------------------------------------------
// WMMA f32 GEMM:  C[M x 128] = A[M x 128] @ W[128 x 128] (+ bias)
// grid = (M/16, 2), block = 128 (4 waves). Each wave: one 16x16 tile.
// Uses V_WMMA_F32_16X16X4_F32, K-loop of 32 steps.
// ---------------------------------------------------------------------------
__global__ __launch_bounds__(128) void gemm_wmma_f32(
        const float* __restrict__ A, const float* __restrict__ W,
        const float* __restrict__ bias, float* __restrict__ C,
        int useBias)
{
    __shared__ float xs[16 * 132];    // A tile, padded (stride 132 kills conflicts)
    __shared__ float ws[128 * 68];    // W slice (64 cols), padded stride 68

    const int tid  = threadIdx.x;
    const int lane = tid & 31;
    const int wave = tid >> 5;                    // 0..3
    const int m0   = blockIdx.x * 16;
    const int nb0  = blockIdx.y * 64;

    // --- stage A tile: 16 consecutive rows => one contiguous 8 KB block
    {
        const float4* srcp = (const float4*)(A + (size_t)m0 * CH);
        #pragma unroll
        for (int i = 0; i < 4; ++i) {
            int idx4 = tid + i * 128;             // 0..511
            float4 v = srcp[idx4];
            int lin = idx4 << 2;
            int r = lin >> 7, c = lin & 127;
            float* d = &xs[r * 132 + c];
            d[0] = v.x; d[1] = v.y; d[2] = v.z; d[3] = v.w;
        }
    }
    // --- stage W slice: rows 0..127, cols nb0..nb0+63
    {
        #pragma unroll
        for (int i = 0; i < 16; ++i) {
            int idx4 = tid + i * 128;             // 0..2047
            int k  = idx4 >> 4;                   // row
            int c4 = idx4 & 15;                   // float4 within row
            float4 v = *(const float4*)(W + (size_t)k * CH + nb0 + (c4 << 2));
            float* d = &ws[k * 68 + (c4 << 2)];
            d[0] = v.x; d[1] = v.y; d[2] = v.z; d[3] = v.w;
        }
    }
    __syncthreads();

    // Fragment addressing per ISA 16x4 A / 4x16 B layouts:
    //   lanes 0-15 carry K={k,k+1}, lanes 16-31 carry K={k+2,k+3}
    const int halfk = (lane >> 4) << 1;   // 0 or 2
    const int arow  = lane & 15;
    const int ncol  = wave * 16 + (lane & 15);

    v8f acc = {};
    #pragma unroll
    for (int k = 0; k < CH; k += 4) {
        v2f a, b;
        a.x = xs[arow * 132 + k + halfk];
        a.y = xs[arow * 132 + k + halfk + 1];
        b.x = ws[(k + halfk) * 68 + ncol];
        b.y = ws[(k + halfk + 1) * 68 + ncol];
        acc = __builtin_amdgcn_wmma_f32_16x16x4_f32(
                  false, a, false, b, (short)0, acc, false, false);
    }

    const float bv = useBias ? bias[nb0 + ncol] : 0.0f;
    // D layout: VGPR r, lanes 0-15 -> M=r, lanes 16-31 -> M=r+8, N=lane&15
    #pragma unroll
    for (int r = 0; r < 8; ++r) {
        int mrow = m0 + r + ((lane >> 4) << 3);
        C[(size_t)mrow * CH + nb0 + ncol] = acc[r] + bv;
    }
}

// ---------------------------------------------------------------------------
// s[i] = <h[i,:], att>   (one wave per node, float4 per lane, shfl reduce)
// ---------------------------------------------------------------------------
__global__ __launch_bounds__(256) void node_score_kernel(
        const float* __restrict__ h, const float* __restrict__ att,
        float* __restrict__ s)
{
    int wid  = (blockIdx.x * blockDim.x + threadIdx.x) >> 5;
    int lane = threadIdx.x & 31;
    if (wid >= NNODES) return;
    float4 hv = *(const float4*)(h + (size_t)wid * CH + (lane << 2));
    float4 av = *(const float4*)(att + (lane << 2));
    float p = hv.x * av.x + hv.y * av.y + hv.z * av.z + hv.w * av.w;
    #pragma unroll
    for (int off = 16; off > 0; off >>= 1) p += __shfl_xor(p, off, 32);
    if (lane == 0) s[wid] = p;
}

// ---------------------------------------------------------------------------
// Edge attention: alpha = leaky_relu(s[dst]+s[src], 0.2); seg-max over src
// ---------------------------------------------------------------------------
__global__ void edge_alpha_kernel(const float* __restrict__ s,
                                  const int* __restrict__ ei,
                                  float* __restrict__ alpha,
                                  float* __restrict__ smax)
{
    int e = blockIdx.x * blockDim.x + threadIdx.x;
    if (e >= ETOT) return;
    int u, v; edge_nodes(e, ei, u, v);
    float a = s[v] + s[u];
    a = (a > 0.0f) ? a : 0.2f * a;
    alpha[e] = a;
    atomicMaxFloat(&smax[u], a);
}

__global__ void edge_exp_kernel(const int* __restrict__ ei,
                                float* __restrict__ alpha,
                                const float* __restrict__ smax,
                                float* __restrict__ ssum)
{
    int e = blockIdx.x * blockDim.x + threadIdx.x;
    if (e >= ETOT) return;
    int u, v; edge_nodes(e, ei, u, v);
    float ex = __expf(alpha[e] - smax[u]);
    alpha[e] = ex;
    atomicAdd(&ssum[u], ex);
}

// out[dst,:] += h[src,:] * (exp / (ssum[src]+eps)); wave per edge
__global__ __launch_bounds__(256) void edge_gt_agg_kernel(
        const float* __restrict__ h, const float* __restrict__ alpha,
        const float* __restrict__ ssum, const int* __restrict__ ei,
        float* __restrict__ out)
{
    int e    = (blockIdx.x * blockDim.x + threadIdx.x) >> 5;
    int lane = threadIdx.x & 31;
    if (e >= ETOT) return;
    int u, v; edge_nodes(e, ei, u, v);
    float coef = alpha[e] / (ssum[u] + EPS_F);
    float4 hv = *(const float4*)(h + (size_t)u * CH + (lane << 2));
    float* o = out + (size_t)v * CH + (lane << 2);
    atomicAdd(o + 0, hv.x * coef);
    atomicAdd(o + 1, hv.y * coef);
    atomicAdd(o + 2, hv.z * coef);
    atomicAdd(o + 3, hv.w * coef);
}

// ---------------------------------------------------------------------------
// GCN pieces
// ---------------------------------------------------------------------------
__global__ void deg_kernel(const int* __restrict__ ei, float* __restrict__ deg) {
    int e = blockIdx.x * blockDim.x + threadIdx.x;
    if (e >= ETOT) return;
    int u, v; edge_nodes(e, ei, u, v);
    atomicAdd(&deg[v], 1.0f);
}

__global__ void dinv_kernel(float* __restrict__ deg) {   // in place deg -> deg^-1/2
    int i = blockIdx.x * blockDim.x + threadIdx.x;
    if (i >= NNODES) return;
    float d = deg[i];
    deg[i] = (d > 0.0f) ? rsqrtf(d) : 0.0f;
}

__global__ __launch_bounds__(256) void edge_gcn_agg_kernel(
        const float* __restrict__ h, const float* __restrict__ dinv,
        const int* __restrict__ ei, float* __restrict__ out)
{
    int e    = (blockIdx.x * blockDim.x + threadIdx.x) >> 5;
    int lane = threadIdx.x & 31;
    if (e >= ETOT) return;
    int u, v; edge_nodes(e, ei, u, v);
    float coef = dinv[u] * dinv[v];
    float4 hv = *(const float4*)(h + (size_t)u * CH + (lane << 2));
    float* o = out + (size_t)v * CH + (lane << 2);
    atomicAdd(o + 0, hv.x * coef);
    atomicAdd(o + 1, hv.y * coef);
    atomicAdd(o + 2, hv.z * coef);
    atomicAdd(o + 3, hv.w * coef);
}

// ---------------------------------------------------------------------------
// elementwise relu (in place)
// ---------------------------------------------------------------------------
__global__ void relu_kernel(float* __restrict__ p, long n) {
    long i = (long)blockIdx.x * blockDim.x + threadIdx.x;
    if (i < n) p[i] = fmaxf(p[i], 0.0f);
}

// ---------------------------------------------------------------------------
// mean-pool over graphs (bias bg folded in); wave per node
// ---------------------------------------------------------------------------
__global__ __launch_bounds__(256) void pool_kernel(
        const float* __restrict__ h, const float* __restrict__ bg,
        const int* __restrict__ batch, float* __restrict__ pooled,
        float* __restrict__ counts)
{
    int i    = (blockIdx.x * blockDim.x + threadIdx.x) >> 5;
    int lane = threadIdx.x & 31;
    if (i >= NNODES) return;
    int g = batch[i];
    float4 hv = *(const float4*)(h + (size_t)i * CH + (lane << 2));
    float4 bv = *(const float4*)(bg + (lane << 2));
    float* p = pooled + (size_t)g * CH + (lane << 2);
    atomicAdd(p + 0, hv.x + bv.x);
    atomicAdd(p + 1, hv.y + bv.y);
    atomicAdd(p + 2, hv.z + bv.z);
    atomicAdd(p + 3, hv.w + bv.w);
    if (lane == 0) atomicAdd(&counts[g], 1.0f);
}

// ---------------------------------------------------------------------------
// head: logits = (pooled/cnt) @ Wfc + bfc, then log_softmax. 1 block, 64 thr.
// ---------------------------------------------------------------------------
__global__ __launch_bounds__(64) void head_kernel(
        const float* __restrict__ pooled, const float* __restrict__ counts,
        const float* __restrict__ Wfc, const float* __restrict__ bfc,
        float* __restrict__ out)
{
    int g = threadIdx.x;                       // 0..63
    float inv = 1.0f / fmaxf(counts[g], 1.0f);
    float logits[OUTC];
    #pragma unroll
    for (int o = 0; o < OUTC; ++o) logits[o] = bfc[o];
    for (int c = 0; c < CH; ++c) {
        float pv = pooled[(size_t)g * CH + c] * inv;
        #pragma unroll
        for (int o = 0; o < OUTC; ++o) logits[o] += pv * Wfc[c * OUTC + o];
    }
    float mx = logits[0];
    #pragma unroll
    for (int o = 1; o < OUTC; ++o) mx = fmaxf(mx, logits[o]);
    float se = 0.0f;
    #pragma unroll
    for (int o = 0; o < OUTC; ++o) se += __expf(logits[o] - mx);
    float lse = mx + __logf(se);
    #pragma unroll
    for (int o = 0; o < OUTC; ++o) out[g * OUTC + o] = logits[o] - lse;
}

// ---------------------------------------------------------------------------
// launcher
// ---------------------------------------------------------------------------
extern "C" void kernel_launch(void* const* d_in, const int* in_sizes, int n_in,
                              void* d_out, int out_size, void* d_ws, size_t ws_size,
                              hipStream_t stream) {
    (void)in_sizes; (void)n_in; (void)out_size; (void)ws_size;

    const float* x    = (const float*)d_in[0];
    const int*   ei   = (const int*)  d_in[1];
    const int*   batch= (const int*)  d_in[2];
    const float* W1   = (const float*)d_in[3];
    const float* b1   = (const float*)d_in[4];
    const float* att1 = (const float*)d_in[5];
    const float* W2   = (const float*)d_in[6];
    const float* b2   = (const float*)d_in[7];
    const float* att2 = (const float*)d_in[8];
    const float* Wg   = (const float*)d_in[9];
    const float* bg   = (const float*)d_in[10];
    const float* Wfc  = (const float*)d_in[11];
    const float* bfc  = (const float*)d_in[12];
    float* out = (float*)d_out;

    // --- workspace layout (floats), 256B-aligned slots
    char* base = (char*)d_ws;
    size_t off = 0;
    auto alloc = [&](size_t nfloats) {
        float* p = (float*)(base + off);
        off += ((nfloats * sizeof(float)) + 255) & ~(size_t)255;
        return p;
    };
    float* hA     = alloc((size_t)NNODES * CH);
    float* hB     = alloc((size_t)NNODES * CH);
    float* s      = alloc(NNODES);
    float* smax   = alloc(NNODES);
    float* ssum   = alloc(NNODES);
    float* alpha  = alloc(ETOT);
    float* deg    = alloc(NNODES);          // becomes dinv in place
    float* pooled = alloc((size_t)NGRAPH * CH);
    float* counts = alloc(NGRAPH);

    const dim3 gemmGrid(NNODES / 16, 2), gemmBlk(128);
    const int nodeBlocks  = (NNODES + 255) / 256;                 // scalar/node
    const int edgeBlocks  = (ETOT + 255) / 256;                   // scalar/edge
    const int nodeWaveBlk = (NNODES * 32) / 256;                  // wave/node (exact)
    const int edgeWaveBlk = (ETOT * 32) / 256;                    // wave/edge (exact)
    const long HN = (long)NNODES * CH;
    const int hBlocks = (int)((HN + 255) / 256);

    auto fill = [&](float* p, float v, long n) {
        fill_kernel<<<(int)((n + 255) / 256), 256, 0, stream>>>(p, v, n);
    };

    // ---------------- GT layer 1 ----------------
    gemm_wmma_f32<<<gemmGrid, gemmBlk, 0, stream>>>(x, W1, b1, hA, 1);
    node_score_kernel<<<nodeWaveBlk, 256, 0, stream>>>(hA, att1, s);
    fill(smax, -INFINITY, NNODES);
    fill(ssum, 0.0f, NNODES);
    edge_alpha_kernel<<<edgeBlocks, 256, 0, stream>>>(s, ei, alpha, smax);
    edge_exp_kernel<<<edgeBlocks, 256, 0, stream>>>(ei, alpha, smax, ssum);
    fill(hB, 0.0f, HN);
    edge_gt_agg_kernel<<<edgeWaveBlk, 256, 0, stream>>>(hA, alpha, ssum, ei, hB);
    relu_kernel<<<hBlocks, 256, 0, stream>>>(hB, HN);

    // ---------------- GT layer 2 ----------------
    gemm_wmma_f32<<<gemmGrid, gemmBlk, 0, stream>>>(hB, W2, b2, hA, 1);
    node_score_kernel<<<nodeWaveBlk, 256, 0, stream>>>(hA, att2, s);
    fill(smax, -INFINITY, NNODES);
    fill(ssum, 0.0f, NNODES);
    edge_alpha_kernel<<<edgeBlocks, 256, 0, stream>>>(s, ei, alpha, smax);
    edge_exp_kernel<<<edgeBlocks, 256, 0, stream>>>(ei, alpha, smax, ssum);
    fill(hB, 0.0f, HN);
    edge_gt_agg_kernel<<<edgeWaveBlk, 256, 0, stream>>>(hA, alpha, ssum, ei, hB);
    relu_kernel<<<hBlocks, 256, 0, stream>>>(hB, HN);

    // ---------------- GCN layer ----------------
    gemm_wmma_f32<<<gemmGrid, gemmBlk, 0, stream>>>(hB, Wg, b1 /*unused*/, hA, 0);
    fill(deg, 0.0f, NNODES);
    deg_kernel<<<edgeBlocks, 256, 0, stream>>>(ei, deg);
    dinv_kernel<<<nodeBlocks, 256, 0, stream>>>(deg);
    fill(hB, 0.0f, HN);
    edge_gcn_agg_kernel<<<edgeWaveBlk, 256, 0, stream>>>(hA, deg, ei, hB);

    // ---------------- pool + head ----------------
    fill(pooled, 0.0f, (long)NGRAPH * CH);
    fill(counts, 0.0f, NGRAPH);
    pool_kernel<<<nodeWaveBlk, 256, 0, stream>>>(hB, bg, batch, pooled, counts);
    head_kernel<<<1, 64, 0, stream>>>(pooled, counts, Wfc, bfc, out);
}